// GRNet_clas_64493228917136
// MI455X (gfx1250) — compile-verified
//
#include <hip/hip_runtime.h>
#include <hip/hip_bf16.h>
#include <math.h>

typedef __attribute__((ext_vector_type(2))) float v2f;
typedef __attribute__((ext_vector_type(8))) float v8f;

// ---------------------------------------------------------------------------
// Stage 1: gridding_reverse (fused with the (B,S^3,3)->(B,3,S,S,S) memory
// reinterpretation) + 4x4x4 maxpool.  Output M3: per batch 12288 floats laid
// out as the flat (3,16,16,16) pooled tensor; reinterpreted later as
// cloud[b, p, c] = M3[b*12288 + p*3 + c].
// ---------------------------------------------------------------------------
__global__ void grid_pool_kernel(const float* __restrict__ data,
                                 float* __restrict__ M3, int B)
{
    int idx = blockIdx.x * blockDim.x + threadIdx.x;
    if (idx >= B * 12288) return;
    int b  = idx / 12288;
    int u  = idx - b * 12288;          // u = c*4096 + pooled cell
    int c  = u >> 12;
    int pc = u & 4095;
    int pi = pc >> 8, pj = (pc >> 4) & 15, pk = pc & 15;
    const float* db = data + (size_t)b * 262144;

    float mx = -3.402823466e+38f;
    for (int ii = 0; ii < 4; ++ii)
      for (int jj = 0; jj < 4; ++jj)
        for (int kk = 0; kk < 4; ++kk) {
            int i = pi * 4 + ii, j = pj * 4 + jj, k = pk * 4 + kk;
            int q    = c * 262144 + i * 4096 + j * 64 + k; // flat idx in (3,64,64,64)
            int vox  = q / 3;
            int comp = q - vox * 3;
            int vx = vox >> 12, vy = (vox >> 6) & 63, vz = vox & 63;
            float basev = (comp == 0) ? (float)vx : (comp == 1) ? (float)vy : (float)vz;
            float wsum = 0.f, pos = 0.f;
            for (int dx = 0; dx < 2; ++dx)
              for (int dy = 0; dy < 2; ++dy)
                for (int dz = 0; dz < 2; ++dz) {
                    int sx = vx + dx - 1, sy = vy + dy - 1, sz = vz + dz - 1;
                    float w = 0.f;
                    if (sx >= 0 && sy >= 0 && sz >= 0) {
                        float dv = db[sx * 4096 + sy * 64 + sz];
                        w = (fabsf(dv) <= 1.0f) ? 1.f : 0.f;
                    }
                    int d = (comp == 0) ? dx : (comp == 1) ? dy : dz;
                    pos  += w * (basev + (float)(d - 1));
                    wsum += w;
                }
            float avg = pos / fmaxf(wsum, 1e-8f);
            float ptv = avg * (1.f / 32.f) - 1.f;
            ptv = (wsum > 0.f) ? ptv : 0.f;
            mx = fmaxf(mx, ptv);
        }
    M3[idx] = mx;
}

// ---------------------------------------------------------------------------
// Stage 2: stable descending top-2048 of 4096 scores (matches jax.lax.top_k).
// ---------------------------------------------------------------------------
__global__ void topk_kernel(const float* __restrict__ M3,
                            const float* __restrict__ noise,
                            int* __restrict__ sel, int B)
{
    __shared__ float s[4096];
    int b = blockIdx.x;
    const float* mb = M3 + (size_t)b * 12288;
    const float* nb = noise + (size_t)b * 4096;
    for (int p = threadIdx.x; p < 4096; p += blockDim.x) {
        float sum = mb[p * 3] + mb[p * 3 + 1] + mb[p * 3 + 2];
        s[p] = (sum != 0.f) ? nb[p] : -1e30f;
    }
    __syncthreads();
    for (int p = threadIdx.x; p < 4096; p += blockDim.x) {
        float sp = s[p];
        int rank = 0;
        for (int j = 0; j < 4096; ++j) {
            float sj = s[j];
            rank += (sj > sp) || (sj == sp && j < p);
        }
        if (rank < 2048) sel[b * 2048 + rank] = p;
    }
}

// ---------------------------------------------------------------------------
// Stage 3: cubic_feature_sampling -> F0 rows of 1792 floats.
// ---------------------------------------------------------------------------
__device__ __forceinline__ int clampi(int v, int lo, int hi)
{ return v < lo ? lo : (v > hi ? hi : v); }

__device__ __forceinline__ void sample_grid(const float* __restrict__ fp,
                                            float* __restrict__ out, int base,
                                            int S, int C,
                                            float px, float py, float pz)
{
    int lx = (int)floorf((px + 1.f) * (0.5f * S));
    int ly = (int)floorf((py + 1.f) * (0.5f * S));
    int lz = (int)floorf((pz + 1.f) * (0.5f * S));
    for (int t = threadIdx.x; t < 8 * C; t += blockDim.x) {
        int corner = t / C, c = t - corner * C;
        int dx = (corner >> 2) & 1, dy = (corner >> 1) & 1, dz = corner & 1;
        int ix = clampi(lx + dx, 0, S - 1);
        int iy = clampi(ly + dy, 0, S - 1);
        int iz = clampi(lz + dz, 0, S - 1);
        out[base + t] = fp[(((size_t)c * S + ix) * S + iy) * S + iz];
    }
}

__global__ void featsample_kernel(const float* __restrict__ M3,
                                  const int* __restrict__ sel,
                                  const float* __restrict__ g32,
                                  const float* __restrict__ g16,
                                  const float* __restrict__ g8,
                                  float* __restrict__ F0)
{
    int row = blockIdx.x;             // b*2048 + r
    int b = row >> 11;
    int p = sel[row];
    const float* mb = M3 + (size_t)b * 12288 + (size_t)p * 3;
    float px = mb[0], py = mb[1], pz = mb[2];
    float* out = F0 + (size_t)row * 1792;
    sample_grid(g32 + (size_t)b * 32 * 32768, out, 0,   32, 32,  px, py, pz);
    sample_grid(g16 + (size_t)b * 64 * 4096,  out, 256, 16, 64,  px, py, pz);
    sample_grid(g8  + (size_t)b * 128 * 512,  out, 768, 8,  128, px, py, pz);
}

// ---------------------------------------------------------------------------
// (K,N) row-major -> (N,K) row-major weight transpose (one-shot per launch).
// ---------------------------------------------------------------------------
__global__ void transpose_kernel(const float* __restrict__ in,
                                 float* __restrict__ out, int K, int N)
{
    long long idx = (long long)blockIdx.x * blockDim.x + threadIdx.x;
    if (idx >= (long long)K * N) return;
    int n = (int)(idx % N);
    long long k = idx / N;
    out[(size_t)n * K + k] = in[idx];
}

// ---------------------------------------------------------------------------
// WMMA fp32 GEMM: C[M,N] = epi(A[M,K] @ Wt^T + bias), Wt is (N,K) row-major.
// One wave per 32(M) x 64(N) tile: 8 accumulators, branch-free K-loop of
// 6 x global_load_b64 + 8 x V_WMMA_F32_16X16X4_F32 per K-step of 4.
// Out-of-range rows/cols are address-CLAMPED (loads stay valid, garbage only
// lands in output rows/cols the guarded epilogue never stores — exact, since
// WMMA output column n depends only on B column n, row m only on A row m).
// Fragment mapping per ISA §7.12.2: lanes 0-15 hold K=k0..k0+1,
// lanes 16-31 hold K=k0+2..k0+3; C/D VGPR r -> rows r / r+8.
// EPI: 0 = bias+relu, 1 = bias+BN+relu, 2 = bias only.
// ---------------------------------------------------------------------------
template <int EPI>
__global__ __launch_bounds__(32) void wmma_gemm(
    const float* __restrict__ A, const float* __restrict__ Wt,
    const float* __restrict__ bias, float* __restrict__ C,
    int M, int N, int K,
    const float* __restrict__ gp, const float* __restrict__ bp,
    const float* __restrict__ mp, const float* __restrict__ vp)
{
    int lane = threadIdx.x;
    int half = lane >> 4;      // 0: K pair k0..k0+1, 1: k0+2..k0+3
    int l15  = lane & 15;
    int tm = blockIdx.y * 32;
    int tn = blockIdx.x * 64;

    int r0 = clampi(tm + l15,      0, M - 1);
    int r1 = clampi(tm + 16 + l15, 0, M - 1);
    const float* pa0 = A + (size_t)r0 * K + half * 2;
    const float* pa1 = A + (size_t)r1 * K + half * 2;
    int c0 = clampi(tn +  0 + l15, 0, N - 1);
    int c1 = clampi(tn + 16 + l15, 0, N - 1);
    int c2 = clampi(tn + 32 + l15, 0, N - 1);
    int c3 = clampi(tn + 48 + l15, 0, N - 1);
    const float* pb0 = Wt + (size_t)c0 * K + half * 2;
    const float* pb1 = Wt + (size_t)c1 * K + half * 2;
    const float* pb2 = Wt + (size_t)c2 * K + half * 2;
    const float* pb3 = Wt + (size_t)c3 * K + half * 2;

    v8f acc[2][4];
#pragma unroll
    for (int rt = 0; rt < 2; ++rt)
#pragma unroll
        for (int j = 0; j < 4; ++j)
#pragma unroll
            for (int i = 0; i < 8; ++i) acc[rt][j][i] = 0.f;

#pragma unroll 2
    for (int k0 = 0; k0 < K; k0 += 4) {
        v2f a0 = *(const v2f*)(pa0 + k0);
        v2f a1 = *(const v2f*)(pa1 + k0);
        v2f b0 = *(const v2f*)(pb0 + k0);
        v2f b1 = *(const v2f*)(pb1 + k0);
        v2f b2 = *(const v2f*)(pb2 + k0);
        v2f b3 = *(const v2f*)(pb3 + k0);
        acc[0][0] = __builtin_amdgcn_wmma_f32_16x16x4_f32(false, a0, false, b0, (short)0, acc[0][0], false, false);
        acc[0][1] = __builtin_amdgcn_wmma_f32_16x16x4_f32(false, a0, false, b1, (short)0, acc[0][1], false, false);
        acc[0][2] = __builtin_amdgcn_wmma_f32_16x16x4_f32(false, a0, false, b2, (short)0, acc[0][2], false, false);
        acc[0][3] = __builtin_amdgcn_wmma_f32_16x16x4_f32(false, a0, false, b3, (short)0, acc[0][3], false, false);
        acc[1][0] = __builtin_amdgcn_wmma_f32_16x16x4_f32(false, a1, false, b0, (short)0, acc[1][0], false, false);
        acc[1][1] = __builtin_amdgcn_wmma_f32_16x16x4_f32(false, a1, false, b1, (short)0, acc[1][1], false, false);
        acc[1][2] = __builtin_amdgcn_wmma_f32_16x16x4_f32(false, a1, false, b2, (short)0, acc[1][2], false, false);
        acc[1][3] = __builtin_amdgcn_wmma_f32_16x16x4_f32(false, a1, false, b3, (short)0, acc[1][3], false, false);
    }

#pragma unroll
    for (int j = 0; j < 4; ++j) {
        int col = tn + j * 16 + l15;
        if (col >= N) continue;
        float bo = bias[col];
        float scale = 1.f, shift = 0.f;
        if (EPI == 1) {
            float inv = rsqrtf(vp[col] + 1e-5f) * gp[col];
            scale = inv;
            shift = bp[col] - mp[col] * inv;
        }
#pragma unroll
        for (int rt = 0; rt < 2; ++rt) {
#pragma unroll
            for (int r = 0; r < 8; ++r) {
                int row = tm + rt * 16 + half * 8 + r;
                if (row < M) {
                    float x = acc[rt][j][r] + bo;
                    if (EPI == 1) x = x * scale + shift;
                    if (EPI != 2) x = fmaxf(x, 0.f);
                    C[(size_t)row * N + col] = x;
                }
            }
        }
    }
}

// ---------------------------------------------------------------------------
// im2col for conv1d (kernel 3, pad 1) over NCH data.
// layout 0: in[b, ch, l] = in[((b*Cin)+ch)*L + l]   (F3: rows are channels)
// layout 1: in[b, ch, l] = in[((b*L)+l)*Cin + ch]   (GEMM outputs, rows=(b,l))
// ---------------------------------------------------------------------------
__global__ void im2col_kernel(const float* __restrict__ in,
                              float* __restrict__ out,
                              int B, int Cin, int L, int layout)
{
    long long idx = (long long)blockIdx.x * blockDim.x + threadIdx.x;
    long long total = (long long)B * L * Cin * 3;
    if (idx >= total) return;
    int K3 = Cin * 3;
    int k = (int)(idx % K3);
    long long rowI = idx / K3;          // b*L + l
    int l = (int)(rowI % L);
    int b = (int)(rowI / L);
    int i = k / 3, t = k - i * 3;
    int ls = l + t - 1;
    float v = 0.f;
    if (ls >= 0 && ls < L)
        v = layout ? in[((size_t)b * L + ls) * Cin + i]
                   : in[((size_t)b * Cin + i) * L + ls];
    out[idx] = v;
}

// (B,32,100) -> h(B,3200):  H[b, o*100+l] = Y4[(b*100+l)*32 + o]
__global__ void repack_kernel(const float* __restrict__ Y4,
                              float* __restrict__ H, int B)
{
    int idx = blockIdx.x * blockDim.x + threadIdx.x;
    if (idx >= B * 3200) return;
    int b = idx / 3200;
    int j = idx - b * 3200;
    int o = j / 100, l = j - o * 100;
    H[idx] = Y4[((size_t)b * 100 + l) * 32 + o];
}

// ---------------------------------------------------------------------------
extern "C" void kernel_launch(void* const* d_in, const int* in_sizes, int n_in,
                              void* d_out, int out_size, void* d_ws, size_t ws_size,
                              hipStream_t stream)
{
    const float* data  = (const float*)d_in[0];
    const float* s32   = (const float*)d_in[1];
    const float* s16   = (const float*)d_in[2];
    const float* s8    = (const float*)d_in[3];
    const float* noise = (const float*)d_in[4];
    const float* w11 = (const float*)d_in[5];  const float* b11 = (const float*)d_in[6];
    const float* w12 = (const float*)d_in[7];  const float* b12 = (const float*)d_in[8];
    const float* w13 = (const float*)d_in[9];  const float* b13 = (const float*)d_in[10];
    const float* c1w = (const float*)d_in[11]; const float* c1b = (const float*)d_in[12];
    const float* g1  = (const float*)d_in[13]; const float* be1 = (const float*)d_in[14];
    const float* m1  = (const float*)d_in[15]; const float* v1  = (const float*)d_in[16];
    const float* c2w = (const float*)d_in[17]; const float* c2b = (const float*)d_in[18];
    const float* g2  = (const float*)d_in[19]; const float* be2 = (const float*)d_in[20];
    const float* m2  = (const float*)d_in[21]; const float* v2  = (const float*)d_in[22];
    const float* c3w = (const float*)d_in[23]; const float* c3b = (const float*)d_in[24];
    const float* g3  = (const float*)d_in[25]; const float* be3 = (const float*)d_in[26];
    const float* m3  = (const float*)d_in[27]; const float* v3  = (const float*)d_in[28];
    const float* c4w = (const float*)d_in[29]; const float* c4b = (const float*)d_in[30];
    const float* g4  = (const float*)d_in[31]; const float* be4 = (const float*)d_in[32];
    const float* m4  = (const float*)d_in[33]; const float* v4  = (const float*)d_in[34];
    const float* w15a = (const float*)d_in[35]; const float* b15a = (const float*)d_in[36];
    const float* w15b = (const float*)d_in[37]; const float* b15b = (const float*)d_in[38];
    const float* w15c = (const float*)d_in[39]; const float* b15c = (const float*)d_in[40];

    int B = in_sizes[0] / 262144;   // (B,1,64,64,64)

    float* ws = (float*)d_ws;
    size_t off = 0;
    auto alloc = [&](size_t n) -> float* {
        float* p = ws + off;
        off += (n + 63) & ~(size_t)63;
        return p;
    };

    float* M3   = alloc((size_t)B * 12288);
    int*   SEL  = (int*)alloc((size_t)B * 2048);
    float* F0   = alloc((size_t)B * 2048 * 1792);
    float* F1   = alloc((size_t)B * 2048 * 1792);
    float* F2   = alloc((size_t)B * 2048 * 448);
    float* F3   = alloc((size_t)B * 2048 * 100);
    float* COL1 = alloc((size_t)B * 100 * 6144);
    float* Y1   = alloc((size_t)B * 100 * 1024);
    float* COL2 = alloc((size_t)B * 100 * 3072);
    float* Y2   = alloc((size_t)B * 100 * 512);
    float* COL3 = alloc((size_t)B * 100 * 1536);
    float* Y3   = alloc((size_t)B * 100 * 128);
    float* COL4 = alloc((size_t)B * 100 * 384);
    float* Y4   = alloc((size_t)B * 100 * 32);
    float* H    = alloc((size_t)B * 3200);
    float* H1   = alloc((size_t)B * 512);
    float* H2   = alloc((size_t)B * 128);
    // transposed (N,K) copies of the row-major (K,N) MLP/FC weights
    float* W11t = alloc((size_t)1792 * 1792);
    float* W12t = alloc((size_t)1792 * 448);
    float* W13t = alloc((size_t)448 * 100);
    float* W15at = alloc((size_t)3200 * 512);
    float* W15bt = alloc((size_t)512 * 128);
    float* W15ct = alloc((size_t)128 * 15);

    dim3 wv(32, 1, 1);
    auto ntile = [](int N) { return (unsigned)((N + 63) / 64); };
    auto mtile = [](int M) { return (unsigned)((M + 31) / 32); };
    auto tgrid = [](long long n) { return (unsigned)((n + 255) / 256); };

    // 0) weight transposes (deterministic, stream-ordered)
    transpose_kernel<<<tgrid(1792LL * 1792), 256, 0, stream>>>(w11, W11t, 1792, 1792);
    transpose_kernel<<<tgrid(1792LL * 448), 256, 0, stream>>>(w12, W12t, 1792, 448);
    transpose_kernel<<<tgrid(448LL * 100), 256, 0, stream>>>(w13, W13t, 448, 100);
    transpose_kernel<<<tgrid(3200LL * 512), 256, 0, stream>>>(w15a, W15at, 3200, 512);
    transpose_kernel<<<tgrid(512LL * 128), 256, 0, stream>>>(w15b, W15bt, 512, 128);
    transpose_kernel<<<tgrid(128LL * 15), 256, 0, stream>>>(w15c, W15ct, 128, 15);

    // 1) gridding_reverse + maxpool
    grid_pool_kernel<<<tgrid((long long)B * 12288), 256, 0, stream>>>(data, M3, B);
    // 2) stable top-2048
    topk_kernel<<<B, 256, 0, stream>>>(M3, noise, SEL, B);
    // 3) cubic feature sampling -> F0 (B*2048, 1792)
    featsample_kernel<<<B * 2048, 256, 0, stream>>>(M3, SEL, s32, s16, s8, F0);

    // 4) point MLP (WMMA fp32 GEMMs, bias+relu)
    int Mr = B * 2048;
    wmma_gemm<0><<<dim3(ntile(1792), mtile(Mr)), wv, 0, stream>>>(F0, W11t, b11, F1, Mr, 1792, 1792,
                                                                  nullptr, nullptr, nullptr, nullptr);
    wmma_gemm<0><<<dim3(ntile(448), mtile(Mr)), wv, 0, stream>>>(F1, W12t, b12, F2, Mr, 448, 1792,
                                                                 nullptr, nullptr, nullptr, nullptr);
    wmma_gemm<0><<<dim3(ntile(100), mtile(Mr)), wv, 0, stream>>>(F2, W13t, b13, F3, Mr, 100, 448,
                                                                 nullptr, nullptr, nullptr, nullptr);

    // 5) conv1d stack as im2col + WMMA GEMM (conv weights are already (O, I*3) = (N,K))
    int Mc = B * 100;
    im2col_kernel<<<tgrid((long long)Mc * 6144), 256, 0, stream>>>(F3, COL1, B, 2048, 100, 0);
    wmma_gemm<1><<<dim3(ntile(1024), mtile(Mc)), wv, 0, stream>>>(COL1, c1w, c1b, Y1, Mc, 1024, 6144,
                                                                  g1, be1, m1, v1);
    im2col_kernel<<<tgrid((long long)Mc * 3072), 256, 0, stream>>>(Y1, COL2, B, 1024, 100, 1);
    wmma_gemm<1><<<dim3(ntile(512), mtile(Mc)), wv, 0, stream>>>(COL2, c2w, c2b, Y2, Mc, 512, 3072,
                                                                 g2, be2, m2, v2);
    im2col_kernel<<<tgrid((long long)Mc * 1536), 256, 0, stream>>>(Y2, COL3, B, 512, 100, 1);
    wmma_gemm<1><<<dim3(ntile(128), mtile(Mc)), wv, 0, stream>>>(COL3, c3w, c3b, Y3, Mc, 128, 1536,
                                                                 g3, be3, m3, v3);
    im2col_kernel<<<tgrid((long long)Mc * 384), 256, 0, stream>>>(Y3, COL4, B, 128, 100, 1);
    wmma_gemm<1><<<dim3(ntile(32), mtile(Mc)), wv, 0, stream>>>(COL4, c4w, c4b, Y4, Mc, 32, 384,
                                                                g4, be4, m4, v4);

    // 6) repack + final FC layers
    repack_kernel<<<tgrid((long long)B * 3200), 256, 0, stream>>>(Y4, H, B);
    wmma_gemm<0><<<dim3(ntile(512), mtile(B)), wv, 0, stream>>>(H, W15at, b15a, H1, B, 512, 3200,
                                                                nullptr, nullptr, nullptr, nullptr);
    wmma_gemm<0><<<dim3(ntile(128), mtile(B)), wv, 0, stream>>>(H1, W15bt, b15b, H2, B, 128, 512,
                                                                nullptr, nullptr, nullptr, nullptr);
    wmma_gemm<2><<<dim3(ntile(15), mtile(B)), wv, 0, stream>>>(H2, W15ct, b15c, (float*)d_out, B, 15, 128,
                                                               nullptr, nullptr, nullptr, nullptr);
}